// NeuralEulerianUpwind_36472862277951
// MI455X (gfx1250) — compile-verified
//
#include <hip/hip_runtime.h>
#include <hip/hip_bf16.h>

#define HH 181
#define WW 360
#define HWN (HH * WW)          // 65160
#define CQ 16                  // num_vels / q channels
#define CH 256                 // hidden_dim
#define NT ((HWN + 15) / 16)   // 4073 column tiles
#define EPI_TOL 1e-5f
#define N_TERMS 19

typedef float v2f __attribute__((ext_vector_type(2)));
typedef float v8f __attribute__((ext_vector_type(8)));

// ---------------------------------------------------------------------------
// Down projection: q[16][HW] = W_down(16x256) @ hidden(256xHW) + b_down
// One wave per 16-column tile; V_WMMA_F32_16X16X4_F32, K-loop of 64 steps.
// ---------------------------------------------------------------------------
__global__ void k_down(const float* __restrict__ hidden,
                       const float* __restrict__ Wd,
                       const float* __restrict__ bd,
                       float* __restrict__ q) {
    int gwave = (blockIdx.x * blockDim.x + threadIdx.x) >> 5;
    int lane  = threadIdx.x & 31;
    if (gwave >= NT) return;                    // uniform per wave
    int p0   = gwave * 16;
    int half = lane >> 4;                        // 0: lanes 0-15, 1: lanes 16-31
    int l    = lane & 15;
    int p    = p0 + l; if (p >= HWN) p = HWN - 1;  // branchless clamp (EXEC stays all-1)

    v8f acc = {0.f,0.f,0.f,0.f,0.f,0.f,0.f,0.f};
    for (int kk = 0; kk < CH / 4; ++kk) {
        int kb = kk * 4 + half * 2;              // A/B K split: lanes<16 -> K 0,1 ; lanes>=16 -> K 2,3
        v2f a, b;
        a.x = Wd[l * CH + kb];                   // A: M=l, 16x4 tile of W_down
        a.y = Wd[l * CH + kb + 1];
        b.x = hidden[(size_t)kb * HWN + p];      // B: K-row striped across lanes (cols)
        b.y = hidden[(size_t)(kb + 1) * HWN + p];
        acc = __builtin_amdgcn_wmma_f32_16x16x4_f32(false, a, false, b,
                                                    (short)0, acc, false, false);
    }
    if (p0 + l < HWN) {
#pragma unroll
        for (int r = 0; r < 8; ++r) {
            int m = half * 8 + r;                // D layout: VGPR r holds row r / r+8
            q[(size_t)m * HWN + p0 + l] = acc[r] + bd[m];
        }
    }
}

// ---------------------------------------------------------------------------
// Upwind RHS stencil (geo-cyclic pad folded into index math)
// ---------------------------------------------------------------------------
__device__ __forceinline__ float rhs_point(const float* __restrict__ qf,
                                           const float* __restrict__ uf,
                                           const float* __restrict__ vf,
                                           float dlat, float dlon,
                                           int c, int h, int w) {
    const float cp[6] = {-2.f/60.f, 15.f/60.f, -60.f/60.f, 20.f/60.f, 30.f/60.f, -3.f/60.f};
    const float cn[6] = { 3.f/60.f,-30.f/60.f, -20.f/60.f, 60.f/60.f,-15.f/60.f,  2.f/60.f};
    const float* qc = qf + (size_t)c * HWN;

    float ql[7];                                  // longitude taps: cyclic wrap
#pragma unroll
    for (int d = -3; d <= 3; ++d) {
        int ww = w + d;
        if (ww < 0) ww += WW; else if (ww >= WW) ww -= WW;
        ql[d + 3] = qc[h * WW + ww];
    }
    float qt[7];                                  // latitude taps: mirror + 180-deg shift
#pragma unroll
    for (int d = -3; d <= 3; ++d) {
        int r = h + d, wc = w;
        if (r < 0)        { r = -1 - r;          wc = w + (WW / 2); if (wc >= WW) wc -= WW; }
        else if (r >= HH) { r = 2 * HH - 1 - r;  wc = w + (WW / 2); if (wc >= WW) wc -= WW; }
        qt[d + 3] = qc[r * WW + wc];
    }
    float dpl = 0.f, dnl = 0.f, dpt = 0.f, dnt = 0.f;
#pragma unroll
    for (int j = 0; j < 6; ++j) {
        dpl += cp[j] * ql[j];  dnl += cn[j] * ql[j + 1];
        dpt += cp[j] * qt[j];  dnt += cn[j] * qt[j + 1];
    }
    float uu = uf[(size_t)c * HWN + h * WW + w];
    float vv = vf[(size_t)c * HWN + h * WW + w];
    float dql = (uu > 0.f ? dpl : dnl) / dlon;
    float dqt = (vv > 0.f ? dpt : dnt) / dlat;
    return -uu * dql - vv * dqt;
}

__global__ void k_rhs_first(const float* __restrict__ q,
                            const float* __restrict__ u,
                            const float* __restrict__ v,
                            const float* __restrict__ dlat_p,
                            const float* __restrict__ dlon_p,
                            float* __restrict__ term,
                            float* __restrict__ result) {
    int idx = blockIdx.x * blockDim.x + threadIdx.x;
    if (idx >= CQ * HWN) return;
    int c = idx / HWN, r = idx - c * HWN, h = r / WW, w = r - h * WW;
    float F = rhs_point(q, u, v, dlat_p[0], dlon_p[0], c, h, w);
    term[idx] = F;
    result[idx] = F;
}

// new_term = rhs(term_in) * dt/(k+1); records global max|new_term|
__global__ void k_rhs_iter(const float* __restrict__ term_in,
                           const float* __restrict__ u,
                           const float* __restrict__ v,
                           const float* __restrict__ dlat_p,
                           const float* __restrict__ dlon_p,
                           const float* __restrict__ dt_p,
                           float kp1,
                           float* __restrict__ term_out,
                           unsigned* __restrict__ maxbits,
                           const int* __restrict__ done) {
    if (*done) return;                            // uniform
    int idx = blockIdx.x * blockDim.x + threadIdx.x;
    bool valid = idx < CQ * HWN;
    float cand = 0.f;
    if (valid) {
        int c = idx / HWN, r = idx - c * HWN, h = r / WW, w = r - h * WW;
        cand = rhs_point(term_in, u, v, dlat_p[0], dlon_p[0], c, h, w) * (dt_p[0] / kp1);
        term_out[idx] = cand;
    }
    float m = valid ? fabsf(cand) : 0.f;          // wave32 max-reduction, one atomic/wave
#pragma unroll
    for (int s = 16; s > 0; s >>= 1) m = fmaxf(m, __shfl_xor(m, s, 32));
    if ((threadIdx.x & 31) == 0) atomicMax(maxbits, __float_as_uint(m));
}

__global__ void k_flag(unsigned* maxbits, int* done, int* add) {
    float m = __uint_as_float(*maxbits);
    if (!*done) {
        if (m < EPI_TOL) { *done = 1; *add = 0; }
        else             { *add = 1; }
    } else {
        *add = 0;
    }
    *maxbits = 0u;
}

__global__ void k_add(const float* __restrict__ term,
                      float* __restrict__ result,
                      const int* __restrict__ add) {
    if (!*add) return;                            // uniform
    int idx = blockIdx.x * blockDim.x + threadIdx.x;
    if (idx < CQ * HWN) result[idx] += term[idx];
}

__global__ void k_init(unsigned* maxbits, int* done, int* add) {
    *maxbits = 0u; *done = 0; *add = 0;
}

// ---------------------------------------------------------------------------
// Up projection with fused epilogue:
// out[256][HW] = W_up(256x16) @ ((q + res*dt)*dw + db) + b_up
// One block (8 waves) per N-tile: the 16x16 dwc tile is computed ONCE into
// LDS, then each wave serves 2 M-tiles (4 WMMAs each), building the shared
// B operand from LDS. Cuts q/res read traffic by 16x vs per-M-tile reloads.
// ---------------------------------------------------------------------------
__global__ void __launch_bounds__(256)
k_up(const float* __restrict__ q,
     const float* __restrict__ res,
     const float* __restrict__ Wu,
     const float* __restrict__ bu,
     const float* __restrict__ dw,
     const float* __restrict__ db,
     const float* __restrict__ dt_p,
     float* __restrict__ out) {
    __shared__ float dwc[CQ * 16];                // 16 channels x 16 columns = 1 KB

    int tile = blockIdx.x;                        // 0..NT-1
    int p0   = tile * 16;
    float dt = dt_p[0];

    // Phase 1: 256 threads compute the dwc tile cooperatively
    {
        int c   = threadIdx.x >> 4;               // channel 0..15
        int col = threadIdx.x & 15;               // column within tile
        int p   = p0 + col; if (p >= HWN) p = HWN - 1;
        size_t g = (size_t)c * HWN + p;
        dwc[c * 16 + col] = (q[g] + res[g] * dt) * dw[c] + db[c];
    }
    __syncthreads();

    int wid  = threadIdx.x >> 5;                  // wave 0..7
    int lane = threadIdx.x & 31;
    int half = lane >> 4, l = lane & 15;

    // Shared B operand (identical for all M-tiles): 4 K-chunks from LDS
    v2f b[4];
#pragma unroll
    for (int kk = 0; kk < 4; ++kk) {
        int kb = kk * 4 + half * 2;
        b[kk].x = dwc[kb * 16 + l];
        b[kk].y = dwc[(kb + 1) * 16 + l];
    }

    bool valid = (p0 + l) < HWN;
#pragma unroll
    for (int t = 0; t < 2; ++t) {
        int mt = wid * 2 + t;                     // M tile 0..15
        v8f acc = {0.f,0.f,0.f,0.f,0.f,0.f,0.f,0.f};
#pragma unroll
        for (int kk = 0; kk < 4; ++kk) {
            int kb = kk * 4 + half * 2;
            v2f a;
            a.x = Wu[(mt * 16 + l) * CQ + kb];
            a.y = Wu[(mt * 16 + l) * CQ + kb + 1];
            acc = __builtin_amdgcn_wmma_f32_16x16x4_f32(false, a, false, b[kk],
                                                        (short)0, acc, false, false);
        }
        if (valid) {
#pragma unroll
            for (int r = 0; r < 8; ++r) {
                int m = mt * 16 + half * 8 + r;
                out[(size_t)m * HWN + p0 + l] = acc[r] + bu[m];
            }
        }
    }
}

// ---------------------------------------------------------------------------
extern "C" void kernel_launch(void* const* d_in, const int* in_sizes, int n_in,
                              void* d_out, int out_size, void* d_ws, size_t ws_size,
                              hipStream_t stream) {
    const float* hidden = (const float*)d_in[0];
    const float* u      = (const float*)d_in[1];
    const float* v      = (const float*)d_in[2];
    const float* dt     = (const float*)d_in[3];
    const float* dlat   = (const float*)d_in[4];
    const float* dlon   = (const float*)d_in[5];
    const float* Wd     = (const float*)d_in[6];
    const float* bd     = (const float*)d_in[7];
    const float* dw     = (const float*)d_in[8];
    const float* db     = (const float*)d_in[9];
    const float* Wu     = (const float*)d_in[10];
    const float* bu     = (const float*)d_in[11];
    float* out = (float*)d_out;

    // workspace layout
    float* q   = (float*)d_ws;
    float* T0  = q  + (size_t)CQ * HWN;
    float* T1  = T0 + (size_t)CQ * HWN;
    float* res = T1 + (size_t)CQ * HWN;
    unsigned* maxbits = (unsigned*)(res + (size_t)CQ * HWN);
    int* done = (int*)(maxbits + 1);
    int* add  = (int*)(maxbits + 2);

    const int N = CQ * HWN;
    const int blocksN = (N + 255) / 256;

    k_init<<<1, 1, 0, stream>>>(maxbits, done, add);

    // down GEMM: NT waves, 8 waves/block
    k_down<<<(NT + 7) / 8, 256, 0, stream>>>(hidden, Wd, bd, q);

    // F_n; result = term = F_n
    k_rhs_first<<<blocksN, 256, 0, stream>>>(q, u, v, dlat, dlon, T0, res);

    float* tin = T0; float* tout = T1;
    for (int k = 1; k <= N_TERMS; ++k) {
        k_rhs_iter<<<blocksN, 256, 0, stream>>>(tin, u, v, dlat, dlon, dt,
                                                (float)(k + 1), tout, maxbits, done);
        k_flag<<<1, 1, 0, stream>>>(maxbits, done, add);
        k_add<<<blocksN, 256, 0, stream>>>(tout, res, add);
        float* t = tin; tin = tout; tout = t;
    }

    // up GEMM, one block per N-tile with LDS-staged dwc
    k_up<<<NT, 256, 0, stream>>>(q, res, Wu, bu, dw, db, dt, out);
}